// Model_38792144617600
// MI455X (gfx1250) — compile-verified
//
#include <hip/hip_runtime.h>

// ---------------------------------------------------------------------------
// 3-layer GCN for MI455X (gfx1250, wave32).
//  - Dense transforms: V_WMMA_F32_16X16X4_F32 (exact fp32 matrix pipe).
//    W is pre-transposed into workspace so both A and B fragments are
//    contiguous float2 loads (global_load_b64); each wave computes two
//    16x16 N-tiles to amortize A loads and run two independent WMMA chains.
//  - Edge aggregation: global_atomic_add_f32 into L2-resident buffers
//    (25.6 MB per feature matrix << 192 MB L2).
// ---------------------------------------------------------------------------

typedef __attribute__((ext_vector_type(2))) float v2f;
typedef __attribute__((ext_vector_type(8))) float v8f;

// ---------------------------------------------------------------------------
// Tiny transpose: WT[n*K + k] = W[k*N + n].  (K*N <= 16384 elements)
// ---------------------------------------------------------------------------
__global__ __launch_bounds__(256) void transpose_w(const float* __restrict__ W,
                                                   float* __restrict__ WT,
                                                   int K, int N) {
  int idx = blockIdx.x * blockDim.x + threadIdx.x;
  if (idx >= K * N) return;
  int k = idx / N;
  int n = idx % N;
  WT[n * K + k] = W[k * N + n];
}

// ---------------------------------------------------------------------------
// GEMM + bias (+ optional ReLU on the input): out[M,N] = relu?(H)[M,128]@W + b
// One wave computes a 16x32 strip (two 16x16 WMMA tiles sharing the A
// fragment). Block = (N/32) waves covering one 16-row M strip.
// M must be a multiple of 16 (50000 = 3125*16).
//
// WMMA f32 16x16x4 operand layouts (ISA 7.12.2):
//   A (16x4): lane l<16 -> row M=l, {K0,K1}; lane l>=16 -> row M=l-16,
//             {K2,K3}  => contiguous float2 per lane.
//   B (4x16): symmetric on columns N=lane&15; with W transposed (WT[n][k])
//             the per-lane pair {B[kb][col], B[kb+1][col]} is contiguous.
//   C/D: VGPR j: lanes 0-15 -> M=j, N=lane; lanes 16-31 -> M=j+8, N=lane-16.
// ---------------------------------------------------------------------------
template <int N, bool RELU>
__global__ __launch_bounds__((N / 32) * 32) void gemm_bias_wmma(
    const float* __restrict__ H, const float* __restrict__ WT,
    const float* __restrict__ bias, float* __restrict__ out) {
  constexpr int K = 128;
  const int lane = threadIdx.x & 31;
  const int wave = threadIdx.x >> 5;
  const int mtile = blockIdx.x;

  const int r     = lane & 15;
  const int klane = (lane >> 4) << 1;  // 0 for lanes 0-15, 2 for lanes 16-31
  const int col0  = (wave * 2 + 0) * 16 + r;
  const int col1  = (wave * 2 + 1) * 16 + r;

  const float* __restrict__ hrow = H + (size_t)(mtile * 16 + r) * K;
  const float* __restrict__ wc0  = WT + (size_t)col0 * K;
  const float* __restrict__ wc1  = WT + (size_t)col1 * K;

  v8f c0 = {};
  v8f c1 = {};
#pragma unroll
  for (int k0 = 0; k0 < K; k0 += 4) {
    const int kb = k0 + klane;
    v2f a = *(const v2f*)(hrow + kb);
    if (RELU) {
      a.x = fmaxf(a.x, 0.0f);
      a.y = fmaxf(a.y, 0.0f);
    }
    const v2f b0 = *(const v2f*)(wc0 + kb);
    const v2f b1 = *(const v2f*)(wc1 + kb);
    // 8 args: (neg_a, A, neg_b, B, c_mod, C, reuse_a, reuse_b)
    c0 = __builtin_amdgcn_wmma_f32_16x16x4_f32(false, a, false, b0, (short)0,
                                               c0, false, false);
    c1 = __builtin_amdgcn_wmma_f32_16x16x4_f32(false, a, false, b1, (short)0,
                                               c1, false, false);
  }

  const float bv0   = bias[col0];
  const float bv1   = bias[col1];
  const int   mbase = mtile * 16 + ((lane >> 4) << 3);  // +8 for upper lanes
#pragma unroll
  for (int j = 0; j < 8; ++j) {
    out[(size_t)(mbase + j) * N + col0] = c0[j] + bv0;
    out[(size_t)(mbase + j) * N + col1] = c1[j] + bv1;
  }
}

// ---------------------------------------------------------------------------
// Edge scatter:  out[dst[e], :] += w[e] * T[src[e], :]
// N/4 threads cooperate on one edge (float4 per thread). Atomic float adds
// (no return -> STOREcnt path) land in L2.
// ---------------------------------------------------------------------------
template <int N>
__global__ __launch_bounds__(256) void scatter_edges(
    const float* __restrict__ T, const float* __restrict__ ew,
    const int* __restrict__ src, const int* __restrict__ dst,
    float* __restrict__ out, int nEdges) {
  constexpr int TPE = N / 4;  // threads per edge
  constexpr int EPB = 256 / TPE;
  const int eLocal = threadIdx.x / TPE;
  const int t      = threadIdx.x % TPE;
  const int e      = blockIdx.x * EPB + eLocal;
  if (e >= nEdges) return;

  const int   s  = src[e];
  const int   d  = dst[e];
  const float wv = ew[e];

  const float4 v = *(const float4*)(T + (size_t)s * N + t * 4);
  float* od      = out + (size_t)d * N + t * 4;
  unsafeAtomicAdd(od + 0, v.x * wv);
  unsafeAtomicAdd(od + 1, v.y * wv);
  unsafeAtomicAdd(od + 2, v.z * wv);
  unsafeAtomicAdd(od + 3, v.w * wv);
}

// ---------------------------------------------------------------------------
// Zero fill (float4 granularity; all our buffers are multiples of 4 floats).
// ---------------------------------------------------------------------------
__global__ __launch_bounds__(256) void zero_f4(float4* __restrict__ p,
                                               size_t n4) {
  size_t i = (size_t)blockIdx.x * blockDim.x + threadIdx.x;
  if (i < n4) p[i] = make_float4(0.f, 0.f, 0.f, 0.f);
}

// ---------------------------------------------------------------------------
extern "C" void kernel_launch(void* const* d_in, const int* in_sizes, int n_in,
                              void* d_out, int out_size, void* d_ws,
                              size_t ws_size, hipStream_t stream) {
  (void)in_sizes; (void)n_in; (void)out_size; (void)ws_size;

  const float* x   = (const float*)d_in[0];
  const float* ew  = (const float*)d_in[1];
  const int*   src = (const int*)d_in[2];
  const int*   dst = (const int*)d_in[3];
  const float* W1  = (const float*)d_in[4];
  const float* b1  = (const float*)d_in[5];
  const float* W2  = (const float*)d_in[6];
  const float* b2  = (const float*)d_in[7];
  const float* W3  = (const float*)d_in[8];
  const float* b3  = (const float*)d_in[9];
  float*       out = (float*)d_out;

  constexpr int M  = 50000;   // nodes (3125 * 16)
  constexpr int E  = 600000;  // edges
  constexpr int MT = M / 16;  // 3125 m-tiles

  float* bufA = (float*)d_ws;            // GEMM outputs (<= 25.6 MB)
  float* bufB = bufA + (size_t)M * 128;  // aggregation buffer (25.6 MB)
  float* WT   = bufB + (size_t)M * 128;  // transposed weights (<= 64 KB)

  const size_t n128_4 = (size_t)M * 128 / 4;
  const size_t n64_4  = (size_t)M * 64 / 4;
  const int zb128 = (int)((n128_4 + 255) / 256);
  const int zb64  = (int)((n64_4 + 255) / 256);

  // ---- Layer 1: t1 = x@W1+b1 ; a1 = scatter(t1) -------------------------
  transpose_w<<<(128 * 128) / 256, 256, 0, stream>>>(W1, WT, 128, 128);
  zero_f4<<<zb128, 256, 0, stream>>>((float4*)bufB, n128_4);
  gemm_bias_wmma<128, false><<<MT, 128, 0, stream>>>(x, WT, b1, bufA);
  scatter_edges<128><<<E / 8, 256, 0, stream>>>(bufA, ew, src, dst, bufB, E);

  // ---- Layer 2: t2 = relu(a1)@W2+b2 ; a2 = scatter(t2) ------------------
  transpose_w<<<(128 * 128) / 256, 256, 0, stream>>>(W2, WT, 128, 128);
  gemm_bias_wmma<128, true><<<MT, 128, 0, stream>>>(bufB, WT, b2, bufA);
  zero_f4<<<zb128, 256, 0, stream>>>((float4*)bufB, n128_4);
  scatter_edges<128><<<E / 8, 256, 0, stream>>>(bufA, ew, src, dst, bufB, E);

  // ---- Layer 3: t3 = relu(a2)@W3+b3 ; out = scatter(t3) -----------------
  transpose_w<<<(128 * 64) / 256, 256, 0, stream>>>(W3, WT, 128, 64);
  gemm_bias_wmma<64, true><<<MT, 64, 0, stream>>>(bufB, WT, b3, bufA);
  zero_f4<<<zb64, 256, 0, stream>>>((float4*)out, n64_4);
  scatter_edges<64><<<E / 16, 256, 0, stream>>>(bufA, ew, src, dst, out, E);
}